// mLSTM_57286273794824
// MI455X (gfx1250) — compile-verified
//
#include <hip/hip_runtime.h>
#include <hip/hip_bf16.h>

#define Bdim 4
#define Tdim 256
#define Hdim 256
#define Ldim 2
#define NWG  16
#define NTHR 256

typedef __attribute__((ext_vector_type(2))) float v2f;
typedef __attribute__((ext_vector_type(4))) float v4f;
typedef __attribute__((ext_vector_type(8))) float v8f;

// ---------------------------------------------------------------------------
// One K-sweep of a 16x16 f32 tile with V_WMMA_F32_16X16X4_F32 (full f32 --
// matches reference precision; these M=4 GEMMs are latency-bound so there is
// no payoff for f16/fp8 down-conversion).
// A rows 4..15 are clamped reads (their D rows are discarded -> EXEC all-1s).
// ---------------------------------------------------------------------------
__device__ __forceinline__ v8f wmma_pass(v8f acc, const float* __restrict__ A,
                                         int strideA, const float* __restrict__ Wn,
                                         int lane) {
  const int row = lane & 15;
  const float* arow = A + (size_t)(row & 3) * strideA;
  const int khalf = (lane >> 4) << 1;  // lanes 0-15: K={0,1}; lanes 16-31: K={2,3}
#pragma unroll 4
  for (int k = 0; k < Hdim; k += 4) {
    v2f a = *(const v2f*)(arow + k + khalf);
    v2f w = *(const v2f*)(Wn + k + khalf);
    acc = __builtin_amdgcn_wmma_f32_16x16x4_f32(false, a, false, w, (short)0, acc,
                                                false, false);
  }
  return acc;
}

// Device-wide sense-reversing barrier over NWG workgroups (co-resident by
// construction: 16 WGs x 8 waves on a many-WGP chip). bar[0]=arrive, bar[1]=gen.
__device__ __forceinline__ void grid_barrier(unsigned* bar) {
  __syncthreads();
  if (threadIdx.x == 0) {
    unsigned gen =
        __hip_atomic_load(&bar[1], __ATOMIC_RELAXED, __HIP_MEMORY_SCOPE_AGENT);
    unsigned prev =
        __hip_atomic_fetch_add(&bar[0], 1u, __ATOMIC_ACQ_REL, __HIP_MEMORY_SCOPE_AGENT);
    if (prev == NWG - 1) {
      __hip_atomic_store(&bar[0], 0u, __ATOMIC_RELAXED, __HIP_MEMORY_SCOPE_AGENT);
      __hip_atomic_fetch_add(&bar[1], 1u, __ATOMIC_RELEASE, __HIP_MEMORY_SCOPE_AGENT);
    } else {
      while (__hip_atomic_load(&bar[1], __ATOMIC_ACQUIRE,
                               __HIP_MEMORY_SCOPE_AGENT) == gen) {
        __builtin_amdgcn_s_sleep(2);
      }
    }
  }
  __syncthreads();
}

__global__ void mlstm_init_kernel(float* __restrict__ ws_gates,
                                  unsigned* __restrict__ bar) {
  if (threadIdx.x < 8) bar[threadIdx.x] = 0u;
  for (int i = threadIdx.x; i < 6 * Bdim * Hdim; i += 64) ws_gates[i] = 0.0f;
}

// ---------------------------------------------------------------------------
// Persistent kernel: entire T x L scan in ONE launch.
//   Phase 1 (per t,l): 96 waves x one 16-col WMMA tile (gates: 2 passes;
//                      q/k/v: 1 pass), activations fused, results -> ws.
//   Phase 2: WG g owns (batch=g&3, 64-col slice=(g>>2)*64). Threads own a
//            float4 column-group x 16-row h-subrange; c RMW is b128/lane,
//            partials LDS-reduced. c lives in d_out's cf region.
// ---------------------------------------------------------------------------
__global__ void __launch_bounds__(NTHR)
mlstm_persistent(const float* __restrict__ x,
                 const float* __restrict__ W_ih, const float* __restrict__ W_hh,
                 const float* __restrict__ bg,
                 const float* __restrict__ Wq, const float* __restrict__ Wk,
                 const float* __restrict__ Wv,
                 const float* __restrict__ bq, const float* __restrict__ bk,
                 const float* __restrict__ bv,
                 float* __restrict__ out, float* __restrict__ hstate,
                 float* __restrict__ cstate,
                 float* __restrict__ ibuf, float* __restrict__ fbuf,
                 float* __restrict__ obuf, float* __restrict__ qbuf,
                 float* __restrict__ kbuf, float* __restrict__ vbuf,
                 unsigned* __restrict__ bar) {
  __shared__ float sq[Hdim], si[Hdim], sf[Hdim], sv[Hdim];
  __shared__ v4f spart[NTHR];

  const int tid  = threadIdx.x;
  const int gtid = blockIdx.x * NTHR + tid;
  const int wave = gtid >> 5;         // 0..127; waves 0..95 own GEMM tiles
  const int lane = tid & 31;
  const int nl   = lane & 15;

  // cell-phase mapping
  const int cb    = blockIdx.x & 3;          // batch
  const int dbase = (blockIdx.x >> 2) << 6;  // 64-wide column slice
  const int dv    = tid & 15;                // float4 column-group in slice
  const int sg    = tid >> 4;                // 0..15, 16-row h-subrange
  const int d4    = dbase + (dv << 2);

  // zero live h/c state (resides in d_out so it ends as hf/cf)
  for (int i = gtid; i < Ldim * Bdim * Hdim + Ldim * Bdim * Hdim * Hdim;
       i += NWG * NTHR)
    hstate[i] = 0.0f;
  grid_barrier(bar);

  for (int t = 0; t < Tdim; ++t) {
    for (int l = 0; l < Ldim; ++l) {
      const float* inpA = (l == 0) ? (x + (size_t)t * Hdim)
                                   : (hstate + (size_t)(l - 1) * Bdim * Hdim);
      const int strideA = (l == 0) ? Tdim * Hdim : Hdim;
      const float* hA    = hstate + (size_t)l * Bdim * Hdim;
      const float* Wih_l = W_ih + (size_t)l * 3 * Hdim * Hdim;
      const float* Whh_l = W_hh + (size_t)l * 3 * Hdim * Hdim;
      const float* b3_l  = bg + (size_t)l * 3 * Hdim;
      const float* Wq_l  = Wq + (size_t)l * Hdim * Hdim;
      const float* Wk_l  = Wk + (size_t)l * Hdim * Hdim;
      const float* Wv_l  = Wv + (size_t)l * Hdim * Hdim;

      // ---- phase 1: WMMA GEMMs -> activated gate / qkv buffers ----
      if (wave < 96) {
        v8f acc = {};
        if (wave < 48) {
          const int ncol0 = wave * 16;
          const int n = ncol0 + nl;
          acc = wmma_pass(acc, inpA, strideA, Wih_l + (size_t)n * Hdim, lane);
          acc = wmma_pass(acc, hA, Hdim, Whh_l + (size_t)n * Hdim, lane);
          const float bias = b3_l[n];
          const int region = ncol0 >> 8;  // 0:i 1:f 2:o
          const int nloc = n & 255;
          float* dest = (region == 0) ? ibuf : (region == 1) ? fbuf : obuf;
#pragma unroll
          for (int j = 0; j < 8; ++j) {
            int m = j + ((lane >> 4) << 3);
            if (m < Bdim) {
              float g = acc[j] + bias;
              float r = (region < 2) ? __expf(g) : 1.0f / (1.0f + __expf(-g));
              dest[m * Hdim + nloc] = r;
            }
          }
        } else {
          const int ncol0 = (wave - 48) * 16;
          const int region = ncol0 >> 8;  // 0:q 1:k 2:v
          const int n = (ncol0 & 255) + nl;
          const float* W  = (region == 0) ? Wq_l : (region == 1) ? Wk_l : Wv_l;
          const float* bb = (region == 0) ? (bq + (size_t)l * Hdim)
                          : (region == 1) ? (bk + (size_t)l * Hdim)
                                          : (bv + (size_t)l * Hdim);
          float* dest = (region == 0) ? qbuf : (region == 1) ? kbuf : vbuf;
          acc = wmma_pass(acc, inpA, strideA, W + (size_t)n * Hdim, lane);
          const float bias = bb[n];
#pragma unroll
          for (int j = 0; j < 8; ++j) {
            int m = j + ((lane >> 4) << 3);
            if (m < Bdim) dest[m * Hdim + n] = acc[j] + bias;
          }
        }
      }
      grid_barrier(bar);

      // ---- phase 2: matrix cell update (b128 RMW) + q.c reduction ----
      sq[tid] = qbuf[cb * Hdim + tid];
      si[tid] = ibuf[cb * Hdim + tid];
      sf[tid] = fbuf[cb * Hdim + tid];
      sv[tid] = vbuf[cb * Hdim + tid];
      __syncthreads();
      {
        const v4f kd4 = *(const v4f*)(kbuf + cb * Hdim + d4);
        float* cbase = cstate + ((size_t)l * Bdim + cb) * Hdim * Hdim;
        v4f acc4 = {};
        const int h0 = sg << 4;
#pragma unroll 4
        for (int h = h0; h < h0 + 16; ++h) {
          v4f cv = *(v4f*)(cbase + h * Hdim + d4);
          const float fh = sf[h];
          const float iv = si[h] * sv[h];
          const float qh = sq[h];
          v4f cn;
          cn.x = fmaf(fh, cv.x, iv * kd4.x);
          cn.y = fmaf(fh, cv.y, iv * kd4.y);
          cn.z = fmaf(fh, cv.z, iv * kd4.z);
          cn.w = fmaf(fh, cv.w, iv * kd4.w);
          *(v4f*)(cbase + h * Hdim + d4) = cn;
          acc4.x = fmaf(qh, cn.x, acc4.x);
          acc4.y = fmaf(qh, cn.y, acc4.y);
          acc4.z = fmaf(qh, cn.z, acc4.z);
          acc4.w = fmaf(qh, cn.w, acc4.w);
        }
        spart[tid] = acc4;  // tid == sg*16 + dv
      }
      __syncthreads();
      if (tid < 16) {
        v4f sum = spart[tid];
#pragma unroll
        for (int j = 1; j < 16; ++j) sum += spart[j * 16 + tid];
        const int dd = dbase + (tid << 2);
        const v4f o4 = *(const v4f*)(obuf + cb * Hdim + dd);
        v4f hn;
        hn.x = o4.x * sum.x;
        hn.y = o4.y * sum.y;
        hn.z = o4.z * sum.z;
        hn.w = o4.w * sum.w;
        *(v4f*)(hstate + (size_t)l * Bdim * Hdim + cb * Hdim + dd) = hn;
        if (l == Ldim - 1)
          *(v4f*)(out + (size_t)cb * Tdim * Hdim + (size_t)t * Hdim + dd) = hn;
      }
      grid_barrier(bar);
    }
  }
}

extern "C" void kernel_launch(void* const* d_in, const int* in_sizes, int n_in,
                              void* d_out, int out_size, void* d_ws, size_t ws_size,
                              hipStream_t stream) {
  const float* x    = (const float*)d_in[0];
  const float* W_ih = (const float*)d_in[1];
  const float* W_hh = (const float*)d_in[2];
  const float* bg   = (const float*)d_in[3];
  const float* Wq   = (const float*)d_in[4];
  const float* Wk   = (const float*)d_in[5];
  const float* Wv   = (const float*)d_in[6];
  const float* bq   = (const float*)d_in[7];
  const float* bk   = (const float*)d_in[8];
  const float* bv   = (const float*)d_in[9];

  // d_out: outputs [B,T,H] | hf [L,B,H] | cf [L,B,H,H]; hf/cf are live state.
  float* out    = (float*)d_out;
  float* hstate = out + (size_t)Bdim * Tdim * Hdim;
  float* cstate = hstate + (size_t)Ldim * Bdim * Hdim;

  float* ws   = (float*)d_ws;
  float* ibuf = ws + 0 * Bdim * Hdim;
  float* fbuf = ws + 1 * Bdim * Hdim;
  float* obuf = ws + 2 * Bdim * Hdim;
  float* qbuf = ws + 3 * Bdim * Hdim;
  float* kbuf = ws + 4 * Bdim * Hdim;
  float* vbuf = ws + 5 * Bdim * Hdim;
  unsigned* bar = (unsigned*)(ws + 6 * Bdim * Hdim);

  mlstm_init_kernel<<<1, 64, 0, stream>>>(ws, bar);
  mlstm_persistent<<<NWG, NTHR, 0, stream>>>(
      x, W_ih, W_hh, bg, Wq, Wk, Wv, bq, bk, bv,
      out, hstate, cstate, ibuf, fbuf, obuf, qbuf, kbuf, vbuf, bar);
}